// Dynamics_2748779069592
// MI455X (gfx1250) — compile-verified
//
#include <hip/hip_runtime.h>

// ---------------------------------------------------------------------------
// 2D periodic diffusion, 16 chunks x 16 steps, overlapped temporal tiling.
//   z <- z + DT*(NU*lap(z) + Q)
// One kernel launch per 16-step chunk; chunk c reads chunk c-1's output slice.
// Tile: 64x64 interior + 16 halo = 96x96 padded, double-buffered in LDS.
// CDNA5 paths: global_load_async_to_lds_b128 (tile fill, ASYNCcnt),
// global_store_async_from_lds_b128 (interior writeback), s_wait_asynccnt.
// ---------------------------------------------------------------------------

#define NX        256
#define TILE      64
#define HALO      16
#define PAD       96            // TILE + 2*HALO
#define LDSS      100           // LDS row stride (mult of 4 for B128, pads banks)
#define THREADS   288           // 96 columns x 3 row-strips (9 waves of 32)
#define STRIP     32            // rows per thread
#define NSTEP     16
#define NCHUNK    16
#define C1        (1.0e-3f * 1.0e-2f)   // DT*NU
#define C2        (1.0e-3f)             // DT

typedef __attribute__((ext_vector_type(4))) int v4i;
typedef __attribute__((address_space(1))) int  as1_int;
typedef __attribute__((address_space(3))) int  as3_int;
typedef __attribute__((address_space(1))) v4i  as1_v4i;
typedef __attribute__((address_space(3))) v4i  as3_v4i;

// ---- async global -> LDS, 16 bytes per lane (ASYNCcnt) --------------------
__device__ __forceinline__ void async_ld16(const float* g, float* l) {
#if __has_builtin(__builtin_amdgcn_global_load_async_to_lds_b128)
  __builtin_amdgcn_global_load_async_to_lds_b128(
      (as1_v4i*)(unsigned long long)g,
      (as3_v4i*)(unsigned)(unsigned long long)l, 0, 0);
#elif __has_builtin(__builtin_amdgcn_global_load_async_to_lds_b32)
#pragma unroll
  for (int j = 0; j < 4; ++j)
    __builtin_amdgcn_global_load_async_to_lds_b32(
        (as1_int*)(unsigned long long)(g + j),
        (as3_int*)(unsigned)(unsigned long long)(l + j), 0, 0);
#else
  unsigned lofs = (unsigned)(unsigned long long)l;
  asm volatile("global_load_async_to_lds_b128 %0, %1, off"
               :: "v"(lofs), "v"(g) : "memory");
#endif
}

// ---- async LDS -> global, 16 bytes per lane (ASYNCcnt) --------------------
__device__ __forceinline__ void async_st16(float* g, const float* l) {
#if __has_builtin(__builtin_amdgcn_global_store_async_from_lds_b128)
  __builtin_amdgcn_global_store_async_from_lds_b128(
      (as1_v4i*)(unsigned long long)g,
      (as3_v4i*)(unsigned)(unsigned long long)l, 0, 0);
#else
  *(float4*)g = *(const float4*)l;   // plain vector copy fallback
#endif
}

__device__ __forceinline__ void wait_async0() {
#if __has_builtin(__builtin_amdgcn_s_wait_asynccnt)
  __builtin_amdgcn_s_wait_asynccnt(0);
#else
  asm volatile("s_wait_asynccnt 0" ::: "memory");
#endif
}

__global__ __launch_bounds__(THREADS, 2)
void diffuse_chunk(const float* __restrict__ src, long long src_bstride,
                   const float* __restrict__ Q, float* __restrict__ out,
                   int chunk) {
  __shared__ float buf[2][PAD * LDSS];

  const int tid = threadIdx.x;
  const int x   = tid % PAD;          // padded-tile column 0..95
  const int yy  = tid / PAD;          // strip id 0..2
  const int b   = blockIdx.y;         // batch element
  const int t   = blockIdx.x;         // tile 0..15
  const int tx0 = (t & 3) * TILE;
  const int ty0 = (t >> 2) * TILE;

  const float* sb = src + (long long)b * src_bstride;

  // ---- async B128 fill of the padded 96x96 tile (wrapped halo) ----------
  // 2304 4-float groups; wrap never splits a group (HALO,TILE mult of 4).
  {
    const int ty  = tid / 24;               // row within a 12-row pass
    const int txg = tid % 24;               // 4-wide column group
    const int x0  = 4 * txg;
    const int gx0 = (tx0 - HALO + x0) & (NX - 1);
#pragma unroll
    for (int i = 0; i < 8; ++i) {
      const int y  = 12 * i + ty;
      const int gy = (ty0 - HALO + y) & (NX - 1);
      async_ld16(sb + gy * NX + gx0, &buf[0][y * LDSS + x0]);
    }
  }
  wait_async0();
  __syncthreads();

  // ---- pull my 32-row column strip + forcing into registers -------------
  const int gx = (tx0 - HALO + x) & (NX - 1);
  float zc[STRIP], q[STRIP];
  const int yb = yy * STRIP;
#pragma unroll
  for (int k = 0; k < STRIP; ++k) {
    const int y  = yb + k;
    const int gy = (ty0 - HALO + y) & (NX - 1);
    zc[k] = buf[0][y * LDSS + x];
    q[k]  = Q[gy * NX + gx];
  }

  const int xl = (x == 0)       ? 0       : x - 1;   // clamp: halo ring is
  const int xr = (x == PAD - 1) ? PAD - 1 : x + 1;   // sacrificial anyway

  // ---- 16 in-LDS steps, double-buffered; 1 barrier/step -----------------
  int cur = 0;
#pragma unroll 1
  for (int s = 0; s < NSTEP; ++s) {
    const float* ob = buf[cur];
    float*       nb = buf[cur ^ 1];
    float prevOld = 0.0f;
#pragma unroll
    for (int k = 0; k < STRIP; ++k) {
      const int y = yb + k;
      const float c  = zc[k];
      const float up = (k == 0)
          ? ob[(y > 0 ? y - 1 : 0) * LDSS + x]           // strip top from LDS
          : prevOld;                                      // my old row above
      const float dn = (k == STRIP - 1)
          ? ob[(y < PAD - 1 ? y + 1 : PAD - 1) * LDSS + x]
          : zc[k + 1];                                    // still holds old
      const float lf = ob[y * LDSS + xl];
      const float rt = ob[y * LDSS + xr];
      const float nv = fmaf(C1, (up + dn + lf + rt - 4.0f * c),
                            fmaf(C2, q[k], c));
      nb[y * LDSS + x] = nv;
      prevOld = c;
      zc[k]   = nv;
    }
    cur ^= 1;
    __syncthreads();
  }

  // ---- async B128 writeback of the valid 64x64 interior from LDS --------
  // buf[cur] holds the final state; loop ends with a barrier so it is
  // complete. 64 rows x 16 groups = 1024 groups over 288 threads.
  {
    float* fin = buf[cur];
    float* ob2 = out + (((long long)b * NCHUNK + chunk) * (NX * NX));
#pragma unroll
    for (int i = 0; i < 4; ++i) {
      const int gidx = tid + THREADS * i;
      if (gidx < TILE * (TILE / 4)) {
        const int y  = HALO + (gidx >> 4);        // 16..79
        const int x0 = HALO + ((gidx & 15) << 2); // 16..76 step 4
        const int ogy = ty0 + (y - HALO);
        const int ogx = tx0 + (x0 - HALO);
        async_st16(ob2 + ogy * NX + ogx, fin + y * LDSS + x0);
      }
    }
  }
  // s_endpgm's implicit wait-idle drains ASYNCcnt before wave termination.
}

extern "C" void kernel_launch(void* const* d_in, const int* in_sizes, int n_in,
                              void* d_out, int out_size, void* d_ws, size_t ws_size,
                              hipStream_t stream) {
  (void)in_sizes; (void)n_in; (void)d_ws; (void)ws_size; (void)out_size;
  const float* inp = (const float*)d_in[0];   // [16, 256, 256]
  const float* Q   = (const float*)d_in[1];   // [256, 256]
  float*       out = (float*)d_out;           // [16, 16, 256, 256]

  for (int c = 0; c < NCHUNK; ++c) {
    const float* src =
        (c == 0) ? inp : (out + (long long)(c - 1) * NX * NX);
    const long long bstride =
        (c == 0) ? (long long)NX * NX : (long long)NCHUNK * NX * NX;
    diffuse_chunk<<<dim3(16, 16), THREADS, 0, stream>>>(src, bstride, Q, out, c);
  }
}